// HungarianMatcher_86852828660470
// MI455X (gfx1250) — compile-verified
//
#include <hip/hip_runtime.h>
#include <cstdint>

// Problem shape (fixed by the reference's setup_inputs()).
constexpr int B = 256;   // batch
constexpr int Q = 256;   // queries (columns of LSAP)
constexpr int T = 128;   // targets (rows of LSAP)
constexpr int D = 3;     // node dim
#define INF_F 1e18f

// ---------------------------------------------------------------------------
// CDNA5 async global->LDS copy (ASYNCcnt path, CDNA5 ISA).
// One 32-bit element per active lane: copied from the global address in vaddr to the LDS address in vdst.
// ---------------------------------------------------------------------------
__device__ __forceinline__ void async_ld_lds_b32(unsigned lds_byte_off,
                                                 const float* gaddr) {
    asm volatile("global_load_async_to_lds_b32 %0, %1, off"
                 :: "v"(lds_byte_off), "v"(gaddr)
                 : "memory");
}
__device__ __forceinline__ void wait_asynccnt0() {
    asm volatile("s_wait_asynccnt 0" ::: "memory");
}
__device__ __forceinline__ unsigned lds_off_of(const void* p) {
    // AS(3)->flat casts put the LDS allocation offset in the low 32 bits.
    return (unsigned)(uintptr_t)p;
}

// ---------------------------------------------------------------------------
// CDNA5 Tensor Data Mover: single-descriptor 1-D DMA of `nelem` f32 from
// global memory into LDS (CDNA5 ISA/§8). Descriptor built in
// SGPR groups s[64:67] (D# group0) and s[72:79] (D# group1); 2-group form
// (VADDR2/VADDR3 = NULL) covers tensors up to 2D. Completion via TENSORcnt.
// ---------------------------------------------------------------------------
__device__ __forceinline__ void tdm_load_1d_f32(unsigned lds_byte_off,
                                                const float* gaddr,
                                                unsigned nelem) {
    const unsigned long long ga = (unsigned long long)(uintptr_t)gaddr;
    // ---- D# group 0 (128b) ----
    const unsigned g0_0 = 1u;                                   // count=1, user desc
    const unsigned g0_1 = lds_byte_off;                         // lds_addr[31:0]
    const unsigned g0_2 = (unsigned)(ga & 0xFFFFFFFFull);       // global_addr[31:0]
    const unsigned g0_3 = (unsigned)((ga >> 32) & 0x1FFFFFFull) // global_addr[56:32]
                        | 0x80000000u;                          // type=2 ("image")
    // ---- D# group 1 (256b) ----
    const unsigned g1_0 = 2u << 16;                 // data_size=2 (4 bytes); no mask
    const unsigned g1_1 = (nelem & 0xFFFFu) << 16;  // tensor_dim0[15:0]  @bits63:48
    const unsigned g1_2 = (nelem >> 16)             // tensor_dim0[31:16] @bits79:64
                        | (1u << 16);               // tensor_dim1=1      @bits111:80
    const unsigned g1_3 = (nelem & 0xFFFFu) << 16;  // tile_dim0          @bits127:112
    const unsigned g1_4 = 0u;                       // tile_dim1=0, tile_dim2=0
    const unsigned g1_5 = nelem;                    // tensor_dim0_stride[31:0]
    const unsigned g1_6 = 0u;                       // stride hi / dim1_stride lo
    const unsigned g1_7 = 0u;                       // tensor_dim1_stride hi
    asm volatile(
        "s_mov_b32 s64, %0\n\t"
        "s_mov_b32 s65, %1\n\t"
        "s_mov_b32 s66, %2\n\t"
        "s_mov_b32 s67, %3\n\t"
        "s_mov_b32 s72, %4\n\t"
        "s_mov_b32 s73, %5\n\t"
        "s_mov_b32 s74, %6\n\t"
        "s_mov_b32 s75, %7\n\t"
        "s_mov_b32 s76, %8\n\t"
        "s_mov_b32 s77, %9\n\t"
        "s_mov_b32 s78, %10\n\t"
        "s_mov_b32 s79, %11\n\t"
        "tensor_load_to_lds s[64:67], s[72:79]"
        :
        : "s"(g0_0), "s"(g0_1), "s"(g0_2), "s"(g0_3),
          "s"(g1_0), "s"(g1_1), "s"(g1_2), "s"(g1_3),
          "s"(g1_4), "s"(g1_5), "s"(g1_6), "s"(g1_7)
        : "s64","s65","s66","s67",
          "s72","s73","s74","s75","s76","s77","s78","s79",
          "memory");
}

// ---------------------------------------------------------------------------
// Kernel 1a: C[b,q,t] = sum_d |pred[b,q,d] - tgt[b,t,d]|   (coalesced over t)
// grid = (Q/2, B), block = 256  -> each block: one batch, two q rows.
// tgt_nodes[b] (T*3 = 384 floats) staged into LDS with per-lane async loads.
// ---------------------------------------------------------------------------
__global__ __launch_bounds__(256) void cost_C_kernel(
        const float* __restrict__ pred, const float* __restrict__ tgt,
        float* __restrict__ C) {
    __shared__ float tg[T * D];  // 384 floats
    const int b   = blockIdx.y;
    const int tid = threadIdx.x;

    const float* gbase = tgt + (size_t)b * T * D;
    for (int i = tid; i < T * D; i += 256)
        async_ld_lds_b32(lds_off_of(&tg[i]), gbase + i);
    wait_asynccnt0();
    __syncthreads();

    const int q = blockIdx.x * 2 + (tid >> 7);
    const int t = tid & (T - 1);
    const float* p = pred + ((size_t)b * Q + q) * D;
    const float p0 = p[0], p1 = p[1], p2 = p[2];
    const float c = fabsf(p0 - tg[t * 3 + 0]) +
                    fabsf(p1 - tg[t * 3 + 1]) +
                    fabsf(p2 - tg[t * 3 + 2]);
    C[((size_t)b * Q + q) * T + t] = c;
}

// ---------------------------------------------------------------------------
// Kernel 1b: CT[b,t,q] (the [T,Q] LSAP cost), coalesced over q.
// grid = (T, B), block = 256 (one q per thread).
// pred_nodes[b] (Q*3 = 768 floats = 3 KB) staged into LDS with a single TDM
// descriptor issued by wave 0 (TDM ignores EXEC; one wave avoids dup DMAs).
// CT (32 MB) stays resident in the 192 MB L2 for the solver.
// ---------------------------------------------------------------------------
__global__ __launch_bounds__(256) void cost_CT_kernel(
        const float* __restrict__ pred, const float* __restrict__ tgt,
        float* __restrict__ CT) {
    __shared__ float pr[Q * D];  // 768 floats, at LDS offset 0
    const int b   = blockIdx.y;
    const int t   = blockIdx.x;
    const int tid = threadIdx.x;

    if (tid < 32) {  // wave 0 issues the DMA and waits on TENSORcnt
        tdm_load_1d_f32(lds_off_of(&pr[0]), pred + (size_t)b * Q * D, Q * D);
        __builtin_amdgcn_s_wait_tensorcnt(0);
    }
    __syncthreads();

    const float* tp = tgt + ((size_t)b * T + t) * D;  // uniform -> scalar loads
    const float t0 = tp[0], t1 = tp[1], t2 = tp[2];
    const float c = fabsf(pr[tid * 3 + 0] - t0) +
                    fabsf(pr[tid * 3 + 1] - t1) +
                    fabsf(pr[tid * 3 + 2] - t2);
    CT[((size_t)b * T + t) * Q + tid] = c;
}

// ---------------------------------------------------------------------------
// Kernel 2: per-batch LSAP (scipy shortest-augmenting-path), one workgroup
// (8 wave32s) per batch, one column per thread, state in LDS.
// Argmin: wave32 shuffle reduction + 8-way cross-wave combine.
// ---------------------------------------------------------------------------
__global__ __launch_bounds__(256) void lsap_kernel(
        const float* __restrict__ CT, float* __restrict__ out) {
    const int b   = blockIdx.x;
    const int tid = threadIdx.x;

    __shared__ float u[T], v[Q], spc[Q];
    __shared__ int   col4row[T], row4col[Q], path[Q];
    __shared__ unsigned char SR[T], SC[Q];
    __shared__ float wmin[8];
    __shared__ int   wjdx[8];
    __shared__ int   s_i, s_sink;
    __shared__ float s_minval;

    if (tid < T) { u[tid] = 0.0f; col4row[tid] = -1; }
    v[tid] = 0.0f; row4col[tid] = -1;
    __syncthreads();

    const float* ctb = CT + (size_t)b * T * Q;

    for (int cur = 0; cur < T; ++cur) {
        // per-row init
        spc[tid] = INF_F; path[tid] = -1; SC[tid] = 0;
        if (tid < T) SR[tid] = 0;
        if (tid == 0) { s_i = cur; s_sink = -1; s_minval = 0.0f; }
        __syncthreads();

        float minVal = 0.0f;
        int   sink   = -1;
        for (;;) {
            const int   i  = s_i;
            const float mv = s_minval;
            const float ui = u[i];
            // reduced cost through row i for this thread's column (L2 hit)
            const float cij = ctb[i * Q + tid];
            const float r   = mv + cij - ui - v[tid];
            if (!SC[tid] && r < spc[tid]) { spc[tid] = r; path[tid] = i; }

            // argmin over masked spc (first-index tie-break, like jnp.argmin)
            float m = SC[tid] ? INF_F : spc[tid];
            int   j = tid;
            #pragma unroll
            for (int off = 16; off > 0; off >>= 1) {
                const float om = __shfl_down(m, off);
                const int   oj = __shfl_down(j, off);
                if (om < m || (om == m && oj < j)) { m = om; j = oj; }
            }
            if ((tid & 31) == 0) { wmin[tid >> 5] = m; wjdx[tid >> 5] = j; }
            __syncthreads();

            if (tid == 0) {
                SR[i] = 1;
                float bm = wmin[0]; int bj = wjdx[0];
                #pragma unroll
                for (int w = 1; w < 8; ++w)
                    if (wmin[w] < bm || (wmin[w] == bm && wjdx[w] < bj)) {
                        bm = wmin[w]; bj = wjdx[w];
                    }
                s_minval = bm;
                SC[bj]   = 1;
                const int rj = row4col[bj];
                if (rj < 0) s_sink = bj; else s_i = rj;
            }
            __syncthreads();
            if (s_sink >= 0) { sink = s_sink; minVal = s_minval; break; }
        }

        // dual updates (with pre-augmentation col4row, as in scipy)
        if (tid < T && SR[tid]) {
            const int   c4 = col4row[tid];
            const float du = (tid == cur) ? minVal
                                          : (minVal - spc[c4 < 0 ? 0 : c4]);
            u[tid] += du;
        }
        if (SC[tid]) v[tid] += spc[tid] - minVal;
        __syncthreads();

        // augment alternating path back from the sink (serial pointer chase)
        if (tid == 0) {
            int j = sink;
            for (;;) {
                const int ii = path[j];
                row4col[j]   = ii;
                const int jn = col4row[ii];
                col4row[ii]  = j;
                if (ii == cur) break;
                j = jn;
            }
        }
        __syncthreads();
    }

    // outputs: row_idx = col4row (matched query per target), col_idx = arange(T)
    const size_t CSZ = (size_t)B * Q * T;
    if (tid < T) {
        out[CSZ + (size_t)b * T + tid]                 = (float)col4row[tid];
        out[CSZ + (size_t)B * T + (size_t)b * T + tid] = (float)tid;
    }
}

// ---------------------------------------------------------------------------
extern "C" void kernel_launch(void* const* d_in, const int* in_sizes, int n_in,
                              void* d_out, int out_size, void* d_ws, size_t ws_size,
                              hipStream_t stream) {
    const float* pred = (const float*)d_in[0];  // [B,Q,3] f32
    const float* tgt  = (const float*)d_in[1];  // [B,T,3] f32
    float* out = (float*)d_out;                 // [C | row_idx | col_idx]
    float* CT  = (float*)d_ws;                  // [B,T,Q] scratch (33.5 MB)

    cost_C_kernel <<<dim3(Q / 2, B), 256, 0, stream>>>(pred, tgt, out);
    cost_CT_kernel<<<dim3(T, B),     256, 0, stream>>>(pred, tgt, CT);
    lsap_kernel   <<<dim3(B),        256, 0, stream>>>(CT, out);
}